// SimpleRNN_80367428043438
// MI455X (gfx1250) — compile-verified
//
#include <hip/hip_runtime.h>

// SimpleRNN: B=8192, T=512, I=4, H=8, O=4 — fp32 WMMA 16x16x4 formulation.
// Each wave handles 16 batch rows; batch rides the WMMA N dimension, weights
// live in the A operand (wave-constant), hidden state hᵀ lives in B-layout.

typedef __attribute__((ext_vector_type(2))) float v2f;
typedef __attribute__((ext_vector_type(8))) float v8f;

#define B_SZ 8192
#define T_SZ 512
#define I_SZ 4
#define H_SZ 8
#define O_SZ 4

// broadcast value held in lanes 0..15 (lane idx) to all lanes
__device__ __forceinline__ float bcast16(float v, int idx) {
    return __int_as_float(__builtin_amdgcn_ds_bpermute(idx << 2, __float_as_int(v)));
}

// tanh(x) = 1 - 2/(exp(2x)+1); v_exp_f32 + v_rcp_f32. Saturates to +-1, NaN-free.
__device__ __forceinline__ float fast_tanh(float x) {
    float e = __builtin_amdgcn_exp2f(x * 2.88539008177792681472f); // 2*log2(e)
    return 1.0f - 2.0f * __builtin_amdgcn_rcpf(e + 1.0f);
}

__global__ __launch_bounds__(128) void rnn_wmma_kernel(
    const float* __restrict__ x,       // [B,T,I]
    const int*   __restrict__ lengths, // [B]
    const float* __restrict__ W_ih,    // [H,I]
    const float* __restrict__ W_hh,    // [H,H]
    const float* __restrict__ b_ih,    // [H]
    const float* __restrict__ b_hh,    // [H]
    const float* __restrict__ W_fc,    // [O,H]
    const float* __restrict__ b_fc,    // [O]
    float*       __restrict__ out)     // [B,T,O]
{
    const int lane = threadIdx.x & 31;
    const int wave = threadIdx.x >> 5;
    const int tile = blockIdx.x * 4 + wave;   // 512 tiles of 16 batch rows
    const int base = tile * 16;

    const int c  = lane & 15;  // column: batch row within tile (N index)
    const int hi = lane >> 4;  // lane half: selects K pair in A/B layout
    const int m  = c;          // A-matrix row index carried by this lane
    const int k0 = 2 * hi;     // first K index held by this lane

    // ---- A operands (wave-constant). 16x4 f32 A layout:
    // lane l (l<16): M=l, {v0,v1} = K={0,1}; lane l+16: M=l, K={2,3}.
    v2f aW, ah1, ah2, afc1, afc2;
    {
        float w0 = W_ih[(m & 7) * I_SZ + k0];
        float w1 = W_ih[(m & 7) * I_SZ + k0 + 1];
        aW.x = (m < H_SZ) ? w0 : 0.0f;
        aW.y = (m < H_SZ) ? w1 : 0.0f;

        float h0 = W_hh[(m & 7) * H_SZ + k0];
        float h1 = W_hh[(m & 7) * H_SZ + k0 + 1];
        float h2 = W_hh[(m & 7) * H_SZ + 4 + k0];
        float h3 = W_hh[(m & 7) * H_SZ + 4 + k0 + 1];
        ah1.x = (m < H_SZ) ? h0 : 0.0f;
        ah1.y = (m < H_SZ) ? h1 : 0.0f;
        ah2.x = (m < H_SZ) ? h2 : 0.0f;
        ah2.y = (m < H_SZ) ? h3 : 0.0f;

        float f0 = W_fc[(m & 3) * H_SZ + k0];
        float f1 = W_fc[(m & 3) * H_SZ + k0 + 1];
        float f2 = W_fc[(m & 3) * H_SZ + 4 + k0];
        float f3 = W_fc[(m & 3) * H_SZ + 4 + k0 + 1];
        afc1.x = (m < O_SZ) ? f0 : 0.0f;
        afc1.y = (m < O_SZ) ? f1 : 0.0f;
        afc2.x = (m < O_SZ) ? f2 : 0.0f;
        afc2.y = (m < O_SZ) ? f3 : 0.0f;
    }

    // ---- C bias operands in 16x16 f32 C/D layout:
    // vgpr r, lanes 0-15: element (M=r, N=lane); lanes 16-31: (M=r+8, N=lane-16).
    // Here M is the hidden/output unit index n, N is the batch column b.
    v8f cbias, cfc;
#pragma unroll
    for (int r = 0; r < 8; ++r) {
        cbias[r] = (hi == 0) ? (b_ih[r] + b_hh[r]) : 0.0f;
        cfc[r]   = (hi == 0 && r < O_SZ) ? b_fc[r] : 0.0f;
    }

    // per-lane x pointer: lane provides xᵀ[k0..k0+1][b=c] each step (B layout)
    const float* xp = x + (size_t)(base + c) * T_SZ * I_SZ + k0;
    float*       op = out + (size_t)(base + c) * T_SZ * O_SZ;
    const int   len = lengths[base + c];

    // hᵀ in B layout (4x16 per chunk): lane l<16: {v0,v1}=h[b=l][{k0,k0+1}] ...
    v2f bh1 = {0.0f, 0.0f};   // K = 0..3 chunk
    v2f bh2 = {0.0f, 0.0f};   // K = 4..7 chunk

    for (int t = 0; t < T_SZ; ++t) {
        // xᵀ tile (4x16, B layout): lane loads x[b, t, k0..k0+1]
        v2f bx = *(const v2f*)xp;
        xp += I_SZ;

        // pre[n][b] = W_ih x + bias  +  W_hh(:,0:4) h(:,0:4)ᵀ  +  W_hh(:,4:8) h(:,4:8)ᵀ
        v8f d = __builtin_amdgcn_wmma_f32_16x16x4_f32(
            false, aW, false, bx, (short)0, cbias, false, false);
        d = __builtin_amdgcn_wmma_f32_16x16x4_f32(
            false, ah1, false, bh1, (short)0, d, false, false);
        d = __builtin_amdgcn_wmma_f32_16x16x4_f32(
            false, ah2, false, bh2, (short)0, d, false, false);

        // h = tanh(pre); rows n>=8 are zero (A rows zero-padded) -> tanh(0)=0
#pragma unroll
        for (int r = 0; r < 8; ++r) d[r] = fast_tanh(d[r]);

        // D layout -> B layout for next step: lanes 16-31 need vgprs 2,3,6,7
        // of lanes 0-15 (uniform 16-lane shift -> ds_bpermute + select).
        float s0 = bcast16(d[2], c);
        float s1 = bcast16(d[3], c);
        float s2 = bcast16(d[6], c);
        float s3 = bcast16(d[7], c);
        bh1.x = hi ? s0 : d[0];
        bh1.y = hi ? s1 : d[1];
        bh2.x = hi ? s2 : d[4];
        bh2.y = hi ? s3 : d[5];

        // FC on length-masked h (recurrence itself continues unmasked)
        bool valid = (t < len);
        v2f bm1 = { valid ? bh1.x : 0.0f, valid ? bh1.y : 0.0f };
        v2f bm2 = { valid ? bh2.x : 0.0f, valid ? bh2.y : 0.0f };

        v8f df = __builtin_amdgcn_wmma_f32_16x16x4_f32(
            false, afc1, false, bm1, (short)0, cfc, false, false);
        df = __builtin_amdgcn_wmma_f32_16x16x4_f32(
            false, afc2, false, bm2, (short)0, df, false, false);

        // out[b][t][0..3] sits in vgprs 0..3 of lanes 0..15 (M=o, N=b)
        if (lane < 16) {
            float4 ov = make_float4(df[0], df[1], df[2], df[3]);
            *(float4*)(op + (size_t)t * O_SZ) = ov;
        }
    }
}

extern "C" void kernel_launch(void* const* d_in, const int* in_sizes, int n_in,
                              void* d_out, int out_size, void* d_ws, size_t ws_size,
                              hipStream_t stream) {
    const float* x       = (const float*)d_in[0];
    const int*   lengths = (const int*)  d_in[1];
    const float* W_ih    = (const float*)d_in[2];
    const float* W_hh    = (const float*)d_in[3];
    const float* b_ih    = (const float*)d_in[4];
    const float* b_hh    = (const float*)d_in[5];
    const float* W_fc    = (const float*)d_in[6];
    const float* b_fc    = (const float*)d_in[7];
    float* out = (float*)d_out;

    // 8192 batch rows / 16 per wave = 512 waves; 4 waves per 128-thread block
    dim3 grid(128), block(128);
    hipLaunchKernelGGL(rnn_wmma_kernel, grid, block, 0, stream,
                       x, lengths, W_ih, W_hh, b_ih, b_hh, W_fc, b_fc, out);
}